// LookingFaceREGNN_47382079209917
// MI455X (gfx1250) — compile-verified
//
#include <hip/hip_runtime.h>
#include <math.h>

// Problem constants (from reference)
constexpr int B  = 32;
constexpr int NF = 64;
constexpr int FD = 64;
constexpr int ND = 256;
constexpr int C  = 8;
constexpr int KTOP = 6;
constexpr int L  = 2;

typedef float v2f __attribute__((ext_vector_type(2)));
typedef float v8f __attribute__((ext_vector_type(8)));
typedef unsigned int u32x4 __attribute__((ext_vector_type(4)));
typedef int i32x4 __attribute__((ext_vector_type(4)));
typedef int i32x8 __attribute__((ext_vector_type(8)));

// One f32 WMMA step: D(16x16) += A(16x4) * B(4x16), full fp32 precision.
// A lane layout: m = lane&15, holds K = 2*(lane>>4) + {0,1}
// B lane layout: n = lane&15, holds K = 2*(lane>>4) + {0,1}
// C/D layout: VGPR j -> row (j + 8*(lane>>4)), col (lane&15)
__device__ __forceinline__ v8f wmma4(v8f acc, v2f a, v2f b) {
    return __builtin_amdgcn_wmma_f32_16x16x4_f32(false, a, false, b,
                                                 (short)0, acc, false, false);
}

__device__ __forceinline__ float wave_sum(float v) {
    for (int o = 16; o; o >>= 1) v += __shfl_xor(v, o);
    return v;
}
__device__ __forceinline__ float wave_max(float v) {
    for (int o = 16; o; o >>= 1) v = fmaxf(v, __shfl_xor(v, o));
    return v;
}

// ---------------------------------------------------------------------------
// Tensor Data Mover: DMA a 2D f32 tile (tile_w x tile_h, row stride
// stride_elems) from global memory into LDS at lds_byte_addr.
// D# built per CDNA5 ISA (groups 0/1; groups 2/3 zero).
// LDS padding: pad_interval=5 (every 64 dwords) + pad_amount=1 (2 dwords)
// -> LDS row stride = tile_w + 2 dwords (bank-conflict-free for 64-wide rows).
// This toolchain's builtin is the 6-arg form:
//   (uint32x4 g0, int32x8 g1, int32x4 g2, int32x4 g3, int32x8 extra, i32 cpol)
// ---------------------------------------------------------------------------
#if defined(__has_builtin)
#if __has_builtin(__builtin_amdgcn_tensor_load_to_lds)
#define HAVE_TDM 1
#endif
#endif

#ifdef HAVE_TDM
__device__ __forceinline__ void tdm_load_tile_f32(unsigned int lds_byte_addr,
                                                  const float* gptr,
                                                  unsigned int tile_w,
                                                  unsigned int tile_h,
                                                  unsigned int stride_elems,
                                                  unsigned int tdim0,
                                                  unsigned int tdim1) {
    const unsigned long long ga = (unsigned long long)(uintptr_t)gptr;
    u32x4 g0;
    g0.x = 1u;                                   // count=1 (valid descriptor)
    g0.y = lds_byte_addr;                        // lds_addr
    g0.z = (unsigned int)ga;                     // global_addr[31:0]
    g0.w = (unsigned int)((ga >> 32) & 0x01FFFFFFu) | 0x80000000u;  // addr[56:32] | type=2
    i32x8 g1;
    g1[0] = (int)((2u << 16) |                   // data_size = 4 bytes
                  (1u << 20) |                   // pad_enable
                  (5u << 22) |                   // pad_interval: every 64 dwords
                  (1u << 25));                   // pad_amount: 2 dwords
    g1[1] = (int)((tdim0 & 0xFFFFu) << 16);                              // tensor_dim0 lo
    g1[2] = (int)(((tdim0 >> 16) & 0xFFFFu) | ((tdim1 & 0xFFFFu) << 16)); // dim0 hi|dim1 lo
    g1[3] = (int)(((tdim1 >> 16) & 0xFFFFu) | ((tile_w & 0xFFFFu) << 16)); // dim1 hi|tile_dim0
    g1[4] = (int)(tile_h & 0xFFFFu);             // tile_dim1 (tile_dim2 = 0)
    g1[5] = (int)stride_elems;                   // tensor_dim0_stride[31:0]
    g1[6] = 0;                                   // stride hi | dim1_stride lo (unused, 2D)
    g1[7] = 0;
    const i32x4 gz4 = {0, 0, 0, 0};
    const i32x8 gz8 = {0, 0, 0, 0, 0, 0, 0, 0};
    __builtin_amdgcn_tensor_load_to_lds(g0, g1, gz4, gz4, gz8, 0);
}
#endif

// ---------------------------------------------------------------------------
// Power-iteration spectral norms.
// blocks [0,64): 1/max(||w1[n]||2,1e-6)      -> norms[n]
// blocks [64,128): 1/max(||w2[n]||2,1e-6)    -> norms[64+n]
// blocks [128,130): 1/(||I+2*scale*Wq Wk^T||2 + 5) -> norms[128+l]
// ---------------------------------------------------------------------------
__global__ void k_norms(const float* __restrict__ w1, const float* __restrict__ w2,
                        const float* __restrict__ gaqkw, float* __restrict__ norms) {
    __shared__ float sv[256];
    __shared__ float su[256];
    __shared__ float sM[64 * 64];
    const int bid = blockIdx.x, tid = threadIdx.x;
    const float* A;
    int R, Cc;
    if (bid < 64) {
        A = w1 + (size_t)bid * 64 * 64; R = 64; Cc = 64;
    } else if (bid < 128) {
        A = w2 + (size_t)(bid - 64) * 64 * 256; R = 64; Cc = 256;
    } else {
        const int l = bid - 128; R = 64; Cc = 64;
        const float* W = gaqkw + (size_t)l * 128 * 64;
        for (int e = tid; e < 64 * 64; e += 256) {
            int i = e >> 6, j = e & 63;
            float s = 0.f;
            for (int t = 0; t < 64; ++t) s += W[i * 64 + t] * W[(64 + j) * 64 + t];
            sM[e] = ((i == j) ? 1.0f : 0.0f) + 2.0f * 0.125f * s;  // scale = NF^-0.5
        }
        __syncthreads();
        A = sM;
    }
    for (int j = tid; j < Cc; j += 256) sv[j] = 1.0f;
    __syncthreads();
    float lastnorm = 1.0f;
    for (int it = 0; it < 64; ++it) {
        for (int i = tid; i < R; i += 256) {
            float s = 0.f;
            for (int j = 0; j < Cc; ++j) s += A[(size_t)i * Cc + j] * sv[j];
            su[i] = s;
        }
        __syncthreads();
        for (int j = tid; j < Cc; j += 256) {
            float s = 0.f;
            for (int i = 0; i < R; ++i) s += A[(size_t)i * Cc + j] * su[i];
            sv[j] = s;
        }
        __syncthreads();
        if (tid == 0) {
            float s = 0.f;
            for (int j = 0; j < Cc; ++j) s += sv[j] * sv[j];
            su[0] = sqrtf(s);
        }
        __syncthreads();
        const float nrm = su[0];
        lastnorm = nrm;
        const float inv = 1.0f / fmaxf(nrm, 1e-30f);
        __syncthreads();
        for (int j = tid; j < Cc; j += 256) sv[j] *= inv;
        __syncthreads();
    }
    if (tid == 0) {
        const float sigma = sqrtf(fmaxf(lastnorm, 0.0f));  // ||A^T A v|| -> lambda_max
        if (bid < 128) norms[bid] = 1.0f / fmaxf(sigma, 1e-6f);
        else           norms[bid] = 1.0f / (sigma + 5.0f);
    }
}

// ---------------------------------------------------------------------------
// h[b,n,j] = gelu(sum_i fused[b,n,i] * w1[n,i,j]*invn1[n] + b1[n,j])
// ---------------------------------------------------------------------------
__global__ void k_mlp1(const float* __restrict__ fused, const float* __restrict__ w1,
                       const float* __restrict__ b1, const float* __restrict__ norms,
                       float* __restrict__ h) {
    const int gw = blockIdx.x * (blockDim.x >> 5) + (threadIdx.x >> 5);
    if (gw >= NF * 8) return;
    const int n = gw >> 3, t = gw & 7, tm = t >> 2, tn = t & 3;
    const int lane = threadIdx.x & 31, mn = lane & 15, kh = (lane >> 4) << 1;
    const float invn = norms[n];
    const float* Arow = fused + ((size_t)(tm * 16 + mn) * NF + n) * FD;
    const float* Bcol = w1 + (size_t)n * FD * FD + (tn * 16 + mn);
    v8f acc = {};
    for (int kk = 0; kk < FD; kk += 4) {
        v2f a, bv;
        a.x = Arow[kk + kh]; a.y = Arow[kk + kh + 1];
        bv.x = Bcol[(size_t)(kk + kh) * FD] * invn;
        bv.y = Bcol[(size_t)(kk + kh + 1) * FD] * invn;
        acc = wmma4(acc, a, bv);
    }
    const int ncol = tn * 16 + (lane & 15);
    const int rb = tm * 16 + ((lane >> 4) << 3);
    const float bias = b1[(size_t)n * FD + ncol];
    for (int j = 0; j < 8; ++j) {
        float v = acc[j] + bias;
        v = 0.5f * v * (1.0f + erff(v * 0.7071067811865475f));  // exact gelu
        h[((size_t)(rb + j) * NF + n) * FD + ncol] = v;
    }
}

// ---------------------------------------------------------------------------
// x[b,o,n] = sum_i h[b,n,i]*w2[n,i,o]*invn2[n] + b2[n,o]   (stored transposed)
// ---------------------------------------------------------------------------
__global__ void k_mlp2(const float* __restrict__ h, const float* __restrict__ w2,
                       const float* __restrict__ b2, const float* __restrict__ norms2,
                       float* __restrict__ x) {
    const int gw = blockIdx.x * (blockDim.x >> 5) + (threadIdx.x >> 5);
    if (gw >= NF * 32) return;
    const int n = gw >> 5, t = gw & 31, tm = t >> 4, tn = t & 15;
    const int lane = threadIdx.x & 31, mn = lane & 15, kh = (lane >> 4) << 1;
    const float invn = norms2[n];
    const float* Arow = h + ((size_t)(tm * 16 + mn) * NF + n) * FD;
    const float* Bcol = w2 + (size_t)n * FD * ND + (tn * 16 + mn);
    v8f acc = {};
    for (int kk = 0; kk < FD; kk += 4) {
        v2f a, bv;
        a.x = Arow[kk + kh]; a.y = Arow[kk + kh + 1];
        bv.x = Bcol[(size_t)(kk + kh) * ND] * invn;
        bv.y = Bcol[(size_t)(kk + kh + 1) * ND] * invn;
        acc = wmma4(acc, a, bv);
    }
    const int ocol = tn * 16 + (lane & 15);
    const int rb = tm * 16 + ((lane >> 4) << 3);
    const float bias = b2[(size_t)n * ND + ocol];
    for (int j = 0; j < 8; ++j)
        x[((size_t)(rb + j) * ND + ocol) * NF + n] = acc[j] + bias;
}

// ---------------------------------------------------------------------------
// qk = x @ edge_qk_w^T ; scatter to q[b,c,d,f] / k[b,c,d,f]
// ---------------------------------------------------------------------------
__global__ void k_qkproj(const float* __restrict__ x, const float* __restrict__ eqk,
                         float* __restrict__ q, float* __restrict__ kbuf) {
    const int gw = blockIdx.x * (blockDim.x >> 5) + (threadIdx.x >> 5);
    if (gw >= 512 * 64) return;
    const int tr = gw >> 6, tn = gw & 63;
    const int lane = threadIdx.x & 31, mn = lane & 15, kh = (lane >> 4) << 1;
    const float* Arow = x + (size_t)(tr * 16 + mn) * NF;
    const float* Brow = eqk + (size_t)(tn * 16 + mn) * NF;  // B[k][col] = eqk[col][k]
    v8f acc = {};
    for (int kk = 0; kk < NF; kk += 4) {
        v2f a, bv;
        a.x = Arow[kk + kh]; a.y = Arow[kk + kh + 1];
        bv.x = Brow[kk + kh]; bv.y = Brow[kk + kh + 1];
        acc = wmma4(acc, a, bv);
    }
    const int r = tn * 16 + (lane & 15);
    const int f = r & 63, c = (r >> 6) & 7, s = r >> 9;
    const int rb = tr * 16 + ((lane >> 4) << 3);
    float* dst = s ? kbuf : q;
    for (int j = 0; j < 8; ++j) {
        const int row = rb + j, b = row >> 8, d = row & 255;
        dst[(((size_t)(b * C + c) * ND) + d) * NF + f] = acc[j];
    }
}

// ---------------------------------------------------------------------------
// S[g,n,m] = scale * sum_f Q[g,n,f]*Kp[g,m,f]   (G matrices, 256x256, K=64)
// ---------------------------------------------------------------------------
__global__ void k_scores(const float* __restrict__ Q, const float* __restrict__ Kp,
                         float* __restrict__ S, int G, float scale) {
    const int gw = blockIdx.x * (blockDim.x >> 5) + (threadIdx.x >> 5);
    if (gw >= G * 256) return;
    const int g = gw >> 8, t = gw & 255, tm = t >> 4, tn = t & 15;
    const int lane = threadIdx.x & 31, mn = lane & 15, kh = (lane >> 4) << 1;
    const float* Arow = Q + ((size_t)g * ND + tm * 16 + mn) * NF;
    const float* Brow = Kp + ((size_t)g * ND + tn * 16 + mn) * NF;
    v8f acc = {};
    for (int kk = 0; kk < NF; kk += 4) {
        v2f a, bv;
        a.x = Arow[kk + kh]; a.y = Arow[kk + kh + 1];
        bv.x = Brow[kk + kh]; bv.y = Brow[kk + kh + 1];
        acc = wmma4(acc, a, bv);
    }
    const int coln = tn * 16 + (lane & 15);
    const int rb = tm * 16 + ((lane >> 4) << 3);
    for (int j = 0; j < 8; ++j)
        S[((size_t)g * ND + rb + j) * ND + coln] = scale * acc[j];
}

// In-place row softmax over last dim (256). One wave per row.
__global__ void k_softmax(float* __restrict__ S, int nrows) {
    const int wid = threadIdx.x >> 5, lane = threadIdx.x & 31;
    const int r = blockIdx.x * (blockDim.x >> 5) + wid;
    if (r >= nrows) return;
    float* p = S + (size_t)r * ND;
    float v[8], mx = -3.4e38f;
    for (int j = 0; j < 8; ++j) { v[j] = p[lane + 32 * j]; mx = fmaxf(mx, v[j]); }
    mx = wave_max(mx);
    float s = 0.f;
    for (int j = 0; j < 8; ++j) { v[j] = expf(v[j] - mx); s += v[j]; }
    s = wave_sum(s);
    const float inv = 1.0f / s;
    for (int j = 0; j < 8; ++j) p[lane + 32 * j] = v[j] * inv;
}

// sum over c: se[b,n,m] = sum_c attn[b,c,n,m]
__global__ void k_sumedge(const float* __restrict__ attn, float* __restrict__ se) {
    const size_t tid = (size_t)blockIdx.x * blockDim.x + threadIdx.x;
    if (tid >= (size_t)B * ND * ND) return;
    const size_t b = tid >> 16, rem = tid & 65535;
    const float* base = attn + b * (size_t)C * 65536 + rem;
    float s = 0.f;
    for (int c = 0; c < C; ++c) s += base[(size_t)c * 65536];
    se[tid] = s;
}

// top-6 indices per row (lower index wins ties, like lax.top_k). One wave/row.
__global__ void k_topk(const float* __restrict__ se, int* __restrict__ topidx) {
    const int wid = threadIdx.x >> 5, lane = threadIdx.x & 31;
    const int r = blockIdx.x * (blockDim.x >> 5) + wid;
    if (r >= B * ND) return;
    const float* p = se + (size_t)r * ND;
    float v[8];
    for (int j = 0; j < 8; ++j) v[j] = p[lane + 32 * j];
    for (int t = 0; t < KTOP; ++t) {
        float m = -3.4e38f; int mi = 0;
        for (int j = 0; j < 8; ++j) {
            const int idx = lane + 32 * j;
            if (v[j] > m) { m = v[j]; mi = idx; }
        }
        for (int o = 16; o; o >>= 1) {
            const float om = __shfl_xor(m, o);
            const int oi = __shfl_xor(mi, o);
            if (om > m || (om == m && oi < mi)) { m = om; mi = oi; }
        }
        if (lane == 0) topidx[(size_t)r * KTOP + t] = mi;
        if (lane == (mi & 31)) v[mi >> 5] = -3.4e38f;
    }
}

// masked row-normalize in place: nr = mask*attn / (rowsum + 1e-6)
__global__ void k_nr(float* __restrict__ attn, const int* __restrict__ topidx) {
    const int wid = threadIdx.x >> 5, lane = threadIdx.x & 31;
    const int r = blockIdx.x * (blockDim.x >> 5) + wid;  // [0, B*C*ND)
    if (r >= B * C * ND) return;
    const int n = r & (ND - 1);
    const int bc = r >> 8;
    const int b = bc >> 3;
    const int* ti = topidx + ((size_t)b * ND + n) * KTOP;
    const int i0 = ti[0], i1 = ti[1], i2 = ti[2], i3 = ti[3], i4 = ti[4], i5 = ti[5];
    float* p = attn + (size_t)r * ND;
    float v[8], s = 0.f;
    for (int j = 0; j < 8; ++j) {
        const int m = lane + 32 * j;
        float t = p[m];
        const bool keep = (m == n) | (m == i0) | (m == i1) | (m == i2) |
                          (m == i3) | (m == i4) | (m == i5);
        t = keep ? t : 0.0f;
        v[j] = t; s += t;
    }
    s = wave_sum(s);
    const float inv = 1.0f / (s + 1e-6f);
    for (int j = 0; j < 8; ++j) p[lane + 32 * j] = v[j] * inv;
}

// colinv[b,c,m] = 1 / (sum_n nr[b,c,n,m] + 1e-6)
__global__ void k_colinv(const float* __restrict__ nr, float* __restrict__ colinv) {
    const int tid = blockIdx.x * blockDim.x + threadIdx.x;
    if (tid >= B * C * ND) return;
    const int bc = tid >> 8, m = tid & 255;
    const float* base = nr + (size_t)bc * 65536 + m;
    float s = 0.f;
    for (int n = 0; n < ND; ++n) s += base[(size_t)n * ND];
    colinv[tid] = 1.0f / (s + 1e-6f);
}

// ---------------------------------------------------------------------------
// edge[g,n,m] = sum_d nr[g,n,d] * nr[g,m,d] * colinv[g,d]   (g = b*C+c)
// The 8.6 GFLOP GEMM. Block = 256 threads (8 waves) -> one 64x64 output tile,
// K chunked 4x64. A/B K-chunks staged into LDS by the Tensor Data Mover
// (row stride padded 64->66 dwords, bank-conflict-free), each wave computes
// two 16x16 tiles sharing one A fragment (2 WMMA per A ds_load pair).
// ---------------------------------------------------------------------------
constexpr int KC = 64;    // K chunk
constexpr int LDST = 66;  // padded LDS row stride in dwords (64 + 2 pad)

__global__ __launch_bounds__(256) void k_edge(const float* __restrict__ nr,
                                              const float* __restrict__ colinv,
                                              float* __restrict__ edge) {
    __shared__ float sA[64 * LDST];
    __shared__ float sB[64 * LDST];
    __shared__ float sCi[KC];
    const int g  = blockIdx.x >> 4;
    const int tb = blockIdx.x & 15;
    const int rb = (tb >> 2) << 6;  // output row block base
    const int cb = (tb & 3) << 6;   // output col block base
    const int wid = threadIdx.x >> 5, lane = threadIdx.x & 31;
    const int tm = wid >> 1;           // 0..3 : row tile within 64x64 block
    const int tn2 = (wid & 1) << 1;    // 0|2  : first of two col tiles
    const int mn = lane & 15, kh = (lane >> 4) << 1;

    const float* ci = colinv + (size_t)g * ND;
    const float* Ag = nr + ((size_t)g * ND + rb) * ND;  // 64 rows x 256
    const float* Bg = nr + ((size_t)g * ND + cb) * ND;  // 64 rows x 256

    v8f acc0 = {}, acc1 = {};
    for (int kc = 0; kc < ND; kc += KC) {
        __syncthreads();  // previous chunk's compute done before overwrite
        if (threadIdx.x < KC) sCi[threadIdx.x] = ci[kc + threadIdx.x];
#ifdef HAVE_TDM
        if (wid == 0) {
            tdm_load_tile_f32((unsigned int)(uintptr_t)sA, Ag + kc, KC, 64, ND, ND, ND);
            tdm_load_tile_f32((unsigned int)(uintptr_t)sB, Bg + kc, KC, 64, ND, ND, ND);
            __builtin_amdgcn_s_wait_tensorcnt(0);
        }
#else
        for (int i = threadIdx.x; i < 64 * KC; i += 256) {
            const int rr = i >> 6, cc2 = i & (KC - 1);
            sA[rr * LDST + cc2] = Ag[(size_t)rr * ND + kc + cc2];
            sB[rr * LDST + cc2] = Bg[(size_t)rr * ND + kc + cc2];
        }
#endif
        __syncthreads();
        const float* Arow = sA + (tm * 16 + mn) * LDST;
        const float* B0 = sB + ((tn2 + 0) * 16 + mn) * LDST;
        const float* B1 = sB + ((tn2 + 1) * 16 + mn) * LDST;
        for (int kk = 0; kk < KC; kk += 4) {
            v2f a, b0, b1;
            a.x = Arow[kk + kh]; a.y = Arow[kk + kh + 1];
            const float c0 = sCi[kk + kh], c1 = sCi[kk + kh + 1];
            b0.x = B0[kk + kh] * c0; b0.y = B0[kk + kh + 1] * c1;
            b1.x = B1[kk + kh] * c0; b1.y = B1[kk + kh + 1] * c1;
            acc0 = wmma4(acc0, a, b0);
            acc1 = wmma4(acc1, a, b1);
        }
    }
    const int rbase = rb + tm * 16 + ((lane >> 4) << 3);
    const int c0 = cb + tn2 * 16 + (lane & 15);
    for (int j = 0; j < 8; ++j) {
        float* row = edge + ((size_t)g * ND + rbase + j) * ND;
        row[c0] = acc0[j];
        row[c0 + 16] = acc1[j];
    }
}

// xs = sigmoid(elu(x))
__global__ void k_elusig(const float* __restrict__ x, float* __restrict__ xs) {
    const int tid = blockIdx.x * blockDim.x + threadIdx.x;
    if (tid >= B * ND * NF) return;
    const float u = x[tid];
    const float e = u > 0.0f ? u : expm1f(u);
    xs[tid] = 1.0f / (1.0f + expf(-e));
}

// qkx = xs @ Wl^T + bl  ->  qg (cols<64), kg (cols>=64)
__global__ void k_gaqk(const float* __restrict__ xs, const float* __restrict__ Wl,
                       const float* __restrict__ bl, float* __restrict__ qg,
                       float* __restrict__ kg) {
    const int gw = blockIdx.x * (blockDim.x >> 5) + (threadIdx.x >> 5);
    if (gw >= 512 * 8) return;
    const int tr = gw >> 3, tn = gw & 7;
    const int lane = threadIdx.x & 31, mn = lane & 15, kh = (lane >> 4) << 1;
    const float* Arow = xs + (size_t)(tr * 16 + mn) * NF;
    const float* Brow = Wl + (size_t)(tn * 16 + mn) * NF;
    v8f acc = {};
    for (int kk = 0; kk < NF; kk += 4) {
        v2f a, bv;
        a.x = Arow[kk + kh]; a.y = Arow[kk + kh + 1];
        bv.x = Brow[kk + kh]; bv.y = Brow[kk + kh + 1];
        acc = wmma4(acc, a, bv);
    }
    const int col = tn * 16 + (lane & 15);
    const int rb = tr * 16 + ((lane >> 4) << 3);
    const float bias = bl[col];
    float* dst = (col < NF) ? (qg + col) : (kg + (col - NF));
    for (int j = 0; j < 8; ++j)
        dst[(size_t)(rb + j) * NF] = acc[j] + bias;
}

// ne[b,n,d] = sum_c (w[c]/wsum) * a[b,n,d]*edge[b,c,n,d] / (rowsum_c + 1e-6)
__global__ void k_ne(const float* __restrict__ a_s, const float* __restrict__ edge,
                     const float* __restrict__ gwv, float* __restrict__ ne) {
    const int wid = threadIdx.x >> 5, lane = threadIdx.x & 31;
    const int r = blockIdx.x * (blockDim.x >> 5) + wid;  // [0, B*ND)
    if (r >= B * ND) return;
    const int b = r >> 8, n = r & 255;
    float wsum = 0.f;
    for (int c = 0; c < C; ++c) wsum += gwv[c];
    const float* ar = a_s + (size_t)r * ND;
    float av[8];
    for (int j = 0; j < 8; ++j) av[j] = ar[lane + 32 * j];
    float acc[8] = {0, 0, 0, 0, 0, 0, 0, 0};
    for (int c = 0; c < C; ++c) {
        const float* er = edge + (((size_t)(b * C + c) * ND) + n) * ND;
        float t[8], s = 0.f;
        for (int j = 0; j < 8; ++j) { t[j] = av[j] * er[lane + 32 * j]; s += t[j]; }
        s = wave_sum(s);
        const float coef = (gwv[c] / wsum) / (s + 1e-6f);
        for (int j = 0; j < 8; ++j) acc[j] += coef * t[j];
    }
    float* o = ne + (size_t)r * ND;
    for (int j = 0; j < 8; ++j) o[lane + 32 * j] = acc[j];
}

// x[b,n,f] += invlip * sum_d ne[b,n,d] * xs[b,d,f]
__global__ void k_upd(const float* __restrict__ ne, const float* __restrict__ xs,
                      const float* __restrict__ invlip, float* __restrict__ x) {
    const int gw = blockIdx.x * (blockDim.x >> 5) + (threadIdx.x >> 5);
    if (gw >= B * 64) return;
    const int b = gw >> 6, t = gw & 63, tm = t >> 2, tn = t & 3;
    const int lane = threadIdx.x & 31, mn = lane & 15, kh = (lane >> 4) << 1;
    const float* Arow = ne + ((size_t)b * ND + tm * 16 + mn) * ND;
    const float* Bbase = xs + (size_t)b * ND * NF + (tn * 16 + mn);
    v8f acc = {};
    for (int kk = 0; kk < ND; kk += 4) {
        v2f a, bv;
        a.x = Arow[kk + kh]; a.y = Arow[kk + kh + 1];
        bv.x = Bbase[(size_t)(kk + kh) * NF];
        bv.y = Bbase[(size_t)(kk + kh + 1) * NF];
        acc = wmma4(acc, a, bv);
    }
    const float il = invlip[0];
    const int col = tn * 16 + (lane & 15);
    const int rb = tm * 16 + ((lane >> 4) << 3);
    for (int j = 0; j < 8; ++j) {
        const size_t idx = ((size_t)b * ND + rb + j) * NF + col;
        x[idx] += acc[j] * il;
    }
}

// ---------------------------------------------------------------------------
extern "C" void kernel_launch(void* const* d_in, const int* in_sizes, int n_in,
                              void* d_out, int out_size, void* d_ws, size_t ws_size,
                              hipStream_t stream) {
    (void)in_sizes; (void)n_in; (void)out_size; (void)ws_size;
    const float* fused = (const float*)d_in[0];
    const float* w1    = (const float*)d_in[1];
    const float* b1    = (const float*)d_in[2];
    const float* w2    = (const float*)d_in[3];
    const float* b2    = (const float*)d_in[4];
    const float* eqk   = (const float*)d_in[5];
    const float* gaqkw = (const float*)d_in[6];
    const float* gaqkb = (const float*)d_in[7];
    const float* gaw   = (const float*)d_in[8];
    float* x = (float*)d_out;  // (B, ND, NF) — built by k_mlp2, updated by k_upd

    float* Wp = (float*)d_ws;
    size_t off = 0;
    auto alloc = [&](size_t n) {
        float* p = Wp + off;
        off += (n + 63) & ~(size_t)63;
        return p;
    };
    float* h      = alloc((size_t)B * NF * FD);
    float* q      = alloc((size_t)B * C * ND * NF);
    float* kbuf   = alloc((size_t)B * C * ND * NF);
    float* attn   = alloc((size_t)B * C * ND * ND);  // becomes nr in place
    float* colinv = alloc((size_t)B * C * ND);
    float* sume   = alloc((size_t)B * ND * ND);
    float* edge   = alloc((size_t)B * C * ND * ND);
    float* xs     = alloc((size_t)B * ND * NF);
    float* a_s    = alloc((size_t)B * ND * ND);
    float* nebuf  = alloc((size_t)B * ND * ND);
    float* qg     = alloc((size_t)B * ND * NF);
    float* kg     = alloc((size_t)B * ND * NF);
    float* norms  = alloc(256);  // [0,64) invn1, [64,128) invn2, [128,130) invlip
    int* topidx   = (int*)alloc((size_t)B * ND * KTOP);

    const float scale = 0.125f;  // NF^-0.5

    k_norms<<<2 * NF + L, 256, 0, stream>>>(w1, w2, gaqkw, norms);
    k_mlp1<<<(NF * 8) / 4, 128, 0, stream>>>(fused, w1, b1, norms, h);
    k_mlp2<<<(NF * 32) / 4, 128, 0, stream>>>(h, w2, b2, norms + NF, x);
    k_qkproj<<<(512 * 64) / 4, 128, 0, stream>>>(x, eqk, q, kbuf);
    k_scores<<<(B * C * 256) / 4, 128, 0, stream>>>(q, kbuf, attn, B * C, scale);
    k_softmax<<<(B * C * ND) / 8, 256, 0, stream>>>(attn, B * C * ND);
    k_sumedge<<<(B * ND * ND) / 256, 256, 0, stream>>>(attn, sume);
    k_topk<<<(B * ND) / 8, 256, 0, stream>>>(sume, topidx);
    k_nr<<<(B * C * ND) / 8, 256, 0, stream>>>(attn, topidx);
    k_colinv<<<(B * C * ND) / 256, 256, 0, stream>>>(attn, colinv);
    k_edge<<<B * C * 16, 256, 0, stream>>>(attn, colinv, edge);

    for (int l = 0; l < L; ++l) {
        k_elusig<<<(B * ND * NF) / 256, 256, 0, stream>>>(x, xs);
        k_gaqk<<<(512 * 8) / 4, 128, 0, stream>>>(
            xs, gaqkw + (size_t)l * 2 * NF * NF, gaqkb + (size_t)l * 2 * NF, qg, kg);
        k_scores<<<(B * 256) / 4, 128, 0, stream>>>(qg, kg, a_s, B, scale);
        k_softmax<<<(B * ND) / 8, 256, 0, stream>>>(a_s, B * ND);
        k_ne<<<(B * ND) / 8, 256, 0, stream>>>(a_s, edge, gaw + (size_t)l * C, nebuf);
        k_upd<<<(B * 64) / 4, 128, 0, stream>>>(nebuf, xs, norms + 2 * NF + l, x);
    }
}